// CapsuleLayer_55456617726616
// MI455X (gfx1250) — compile-verified
//
#include <hip/hip_runtime.h>
#include <hip/hip_bf16.h>
#include <math.h>

// CapsNet dynamic routing for MI455X (gfx1250, wave32).
//
// Plan (see roofline in commit message):
//  Kernel 1: u_hat[b,j,i,d] = sum_k u[b,i,k]*W[j,i,d,k] via V_WMMA_F32_16X16X4_F32.
//            Per (i, 16-batch tile) one wave does, for each j: two chained K=4
//            fp32 WMMAs (M=16 batch, N=16 d, K=8 split into 0..3 / 4..7).
//            Result (189 MB) goes to d_ws — resident in MI455X's 192 MB L2.
//  Kernel 2: fused 3-iteration routing. One block per batch sample b,
//            320 threads = 10 wave32s = one wave per output capsule j.
//            b_ij + softmax stats in LDS; u_hat re-read from L2.

#define CAPS_B   256
#define CAPS_NI  1152
#define CAPS_DI  8
#define CAPS_NJ  10
#define CAPS_DJ  16
#define CAPS_EPS 1e-7f

typedef __attribute__((ext_vector_type(2))) float v2f;
typedef __attribute__((ext_vector_type(8))) float v8f;

// ---------------------------------------------------------------------------
// Kernel 1: u_hat GEMM with fp32 WMMA.
// Grid: 2304 blocks x 256 threads (8 waves). wave units = NI * (B/16) = 18432.
// A (16x4 fp32) layout: VGPR0 = {K=0 | K=2}, VGPR1 = {K=1 | K=3}; lane&15 = M.
// B (4x16 fp32) layout: VGPR0 = {K=0 | K=2}, VGPR1 = {K=1 | K=3}; lane&15 = N.
// D (16x16 fp32): VGPR v holds (M = v + (lane>=16 ? 8 : 0), N = lane&15).
// EXEC is all-ones everywhere before/at the WMMAs (no divergence).
// ---------------------------------------------------------------------------
__global__ __launch_bounds__(256)
void caps_uhat_wmma_kernel(const float* __restrict__ u,
                           const float* __restrict__ W,
                           float* __restrict__ uhat)
{
    const int wave = blockIdx.x * (blockDim.x >> 5) + (threadIdx.x >> 5);
    const int lane = threadIdx.x & 31;

    const int i  = wave % CAPS_NI;      // input capsule index
    const int bt = wave / CAPS_NI;      // batch tile (0..15), 16 samples each

    const int row    = lane & 15;        // M (batch) for A; N (=d) for B/D
    const int khalf  = (lane >> 4) * 2;  // 0 (lanes 0-15) or 2 (lanes 16-31)
    const int m_base = (lane >> 4) * 8;  // D-matrix M offset for this lane half

    // A operand: u[b, i, k], b = bt*16 + row. Loaded once, reused for all j.
    const float* up = u + ((size_t)(bt * 16 + row) * CAPS_NI + i) * CAPS_DI;
    v2f a0, a1;
    a0.x = up[khalf + 0]; a0.y = up[khalf + 1];   // K = 0..3 block
    a1.x = up[khalf + 4]; a1.y = up[khalf + 5];   // K = 4..7 block

    for (int j = 0; j < CAPS_NJ; ++j) {
        // B operand: W[j, i, d=row, k]; last dim k is contiguous -> b64 load.
        const float* wp = W + (((size_t)j * CAPS_NI + i) * CAPS_DJ + row) * CAPS_DI;
        v2f b0, b1;
        b0.x = wp[khalf + 0]; b0.y = wp[khalf + 1];
        b1.x = wp[khalf + 4]; b1.y = wp[khalf + 5];

        v8f c = {};
        c = __builtin_amdgcn_wmma_f32_16x16x4_f32(false, a0, false, b0,
                                                  (short)0, c, false, false);
        c = __builtin_amdgcn_wmma_f32_16x16x4_f32(false, a1, false, b1,
                                                  (short)0, c, false, false);

        // Store D: element (b = bt*16 + m_base + v, d = row).
        // Consecutive lanes -> consecutive d -> coalesced 64B per half-wave.
#pragma unroll
        for (int v = 0; v < 8; ++v) {
            const int b_idx = bt * 16 + m_base + v;
            const size_t off =
                (((size_t)b_idx * CAPS_NJ + j) * CAPS_NI + i) * CAPS_DJ + row;
            uhat[off] = c[v];
        }
    }
}

// ---------------------------------------------------------------------------
// Kernel 2: fused dynamic routing. One block per batch sample.
// 320 threads = 10 wave32s; wave j owns output capsule j.
// LDS: bb (46KB) + mx/invden (9.2KB) + sv (640B) < 64KB static.
// ---------------------------------------------------------------------------
__global__ __launch_bounds__(320)
void caps_routing_kernel(const float* __restrict__ uhat,
                         float* __restrict__ out)
{
    __shared__ float bb[CAPS_NJ][CAPS_NI];   // routing logits b_ij
    __shared__ float mx[CAPS_NI];            // per-i softmax max over j
    __shared__ float invden[CAPS_NI];        // per-i softmax 1/sum over j
    __shared__ float sv[CAPS_NJ][CAPS_DJ];   // s_j then v_j

    const int b    = blockIdx.x;
    const int tid  = threadIdx.x;
    const int wv   = tid >> 5;               // wave id = capsule j
    const int lane = tid & 31;

    for (int t = tid; t < CAPS_NJ * CAPS_NI; t += 320)
        (&bb[0][0])[t] = 0.0f;
    __syncthreads();

    const float* base = uhat + (size_t)b * CAPS_NJ * CAPS_NI * CAPS_DJ;

    for (int it = 0; it < 3; ++it) {
        // ---- softmax stats over j (per input capsule i) ----
        for (int i = tid; i < CAPS_NI; i += 320) {
            float m = -3.402823e38f;
            for (int j = 0; j < CAPS_NJ; ++j) m = fmaxf(m, bb[j][i]);
            float s = 0.0f;
            for (int j = 0; j < CAPS_NJ; ++j) s += __expf(bb[j][i] - m);
            mx[i] = m;
            invden[i] = 1.0f / s;
        }
        __syncthreads();

        // ---- s_j = sum_i c_ij * u_hat[b,j,i,:]  (wave j, lanes sweep i) ----
        if (wv < CAPS_NJ) {
            const int j = wv;
            float acc[CAPS_DJ];
#pragma unroll
            for (int d = 0; d < CAPS_DJ; ++d) acc[d] = 0.0f;

            const float* uj = base + (size_t)j * CAPS_NI * CAPS_DJ;
            for (int i = lane; i < CAPS_NI; i += 32) {
                const float c = __expf(bb[j][i] - mx[i]) * invden[i];
                const float4* p = (const float4*)(uj + (size_t)i * CAPS_DJ);
#pragma unroll
                for (int q = 0; q < 4; ++q) {
                    float4 v4 = p[q];
                    acc[q * 4 + 0] = fmaf(c, v4.x, acc[q * 4 + 0]);
                    acc[q * 4 + 1] = fmaf(c, v4.y, acc[q * 4 + 1]);
                    acc[q * 4 + 2] = fmaf(c, v4.z, acc[q * 4 + 2]);
                    acc[q * 4 + 3] = fmaf(c, v4.w, acc[q * 4 + 3]);
                }
            }
            // wave32 cross-lane reduction
#pragma unroll
            for (int d = 0; d < CAPS_DJ; ++d)
#pragma unroll
                for (int off = 16; off > 0; off >>= 1)
                    acc[d] += __shfl_xor(acc[d], off, 32);
            if (lane == 0)
                for (int d = 0; d < CAPS_DJ; ++d) sv[wv][d] = acc[d];
        }
        __syncthreads();

        // ---- squash ----
        if (tid < CAPS_NJ) {
            float sq = 0.0f;
            for (int d = 0; d < CAPS_DJ; ++d) {
                float x = sv[tid][d];
                sq += x * x;
            }
            const float scale = (sq / (1.0f + sq)) * rsqrtf(sq + CAPS_EPS);
            for (int d = 0; d < CAPS_DJ; ++d) sv[tid][d] *= scale;
        }
        __syncthreads();

        // ---- agreement update (skipped on last iteration) ----
        if (it < 2 && wv < CAPS_NJ) {
            const int j = wv;
            float v[CAPS_DJ];
#pragma unroll
            for (int d = 0; d < CAPS_DJ; ++d) v[d] = sv[j][d];

            const float* uj = base + (size_t)j * CAPS_NI * CAPS_DJ;
            for (int i = lane; i < CAPS_NI; i += 32) {
                const float4* p = (const float4*)(uj + (size_t)i * CAPS_DJ);
                float dot = 0.0f;
#pragma unroll
                for (int q = 0; q < 4; ++q) {
                    float4 v4 = p[q];
                    dot = fmaf(v[q * 4 + 0], v4.x, dot);
                    dot = fmaf(v[q * 4 + 1], v4.y, dot);
                    dot = fmaf(v[q * 4 + 2], v4.z, dot);
                    dot = fmaf(v[q * 4 + 3], v4.w, dot);
                }
                bb[j][i] += dot;
            }
        }
        __syncthreads();
    }

    // ---- write v_j: out[b, j, d] ----
    if (tid < CAPS_NJ * CAPS_DJ)
        out[(size_t)b * CAPS_NJ * CAPS_DJ + tid] = (&sv[0][0])[tid];
}

extern "C" void kernel_launch(void* const* d_in, const int* in_sizes, int n_in,
                              void* d_out, int out_size, void* d_ws, size_t ws_size,
                              hipStream_t stream) {
    const float* u = (const float*)d_in[0];   // (256, 1152, 8) fp32
    const float* W = (const float*)d_in[1];   // (10, 1152, 16, 8) fp32
    float* out  = (float*)d_out;              // (256, 10, 16) fp32
    float* uhat = (float*)d_ws;               // needs 256*10*1152*16*4 = 189 MB

    // Kernel 1: 18432 wave-units (NI * B/16), 8 waves per 256-thread block.
    caps_uhat_wmma_kernel<<<(CAPS_NI * (CAPS_B / 16)) / 8, 256, 0, stream>>>(u, W, uhat);

    // Kernel 2: one block per batch sample, 10 wave32s per block.
    caps_routing_kernel<<<CAPS_B, 320, 0, stream>>>(uhat, out);
}